// WindowAttention_36971078484564
// MI455X (gfx1250) — compile-verified
//
#include <hip/hip_runtime.h>
#include <math.h>

typedef float v2f __attribute__((ext_vector_type(2)));
typedef float v8f __attribute__((ext_vector_type(8)));

#define DIMC     96
#define HEADS    3
#define HEAD_DIM 32
#define NTOK     64
#define NWIN     8192
#define QKV_ROW  288              // 3*DIM
#define ATT_SCALE 0.17677669529663687f   // 32^-0.5

// ---------------------------------------------------------------------------
// Kernel A: relative-position bias MLP.  [64,64,2] -> relu(.@w1+b1) -> @w2+b2
// Output layout: bias[h][i][j], h in 0..2, (i,j) in 64x64.
// ---------------------------------------------------------------------------
__global__ __launch_bounds__(256) void winattn_bias_kernel(
    const float* __restrict__ rel_pos,  // [64*64, 2]
    const float* __restrict__ w1,       // [2, 256]
    const float* __restrict__ b1,       // [256]
    const float* __restrict__ w2,       // [256, 3]
    const float* __restrict__ b2,       // [3]
    float* __restrict__ bias)           // [3, 64, 64]
{
    int idx = blockIdx.x * blockDim.x + threadIdx.x;   // 0..4095
    if (idx >= NTOK * NTOK) return;
    float x0 = rel_pos[idx * 2 + 0];
    float x1 = rel_pos[idx * 2 + 1];
    float a0 = b2[0], a1 = b2[1], a2 = b2[2];
    #pragma unroll 4
    for (int k = 0; k < 256; ++k) {
        float hk = fmaxf(0.0f, fmaf(x0, w1[k], fmaf(x1, w1[256 + k], b1[k])));
        a0 = fmaf(hk, w2[k * 3 + 0], a0);
        a1 = fmaf(hk, w2[k * 3 + 1], a1);
        a2 = fmaf(hk, w2[k * 3 + 2], a2);
    }
    bias[0 * 4096 + idx] = a0;
    bias[1 * 4096 + idx] = a1;
    bias[2 * 4096 + idx] = a2;
}

// ---------------------------------------------------------------------------
// Kernel B: one block = (branch, window, head). 4 waves; wave w owns rows
// m0=16w..16w+15 of the 64x64 attention for its head. fp32 WMMA 16x16x4.
// ---------------------------------------------------------------------------
#define KV_STRIDE 36    // padded LDS row stride (floats): bank-conflict free, 16B aligned
#define P_STRIDE  68

__global__ __launch_bounds__(128) void winattn_attn_kernel(
    const float* __restrict__ qkv0,
    const float* __restrict__ qkv1,
    const float* __restrict__ bias,     // [3,64,64]
    float* __restrict__ out)            // [2][8192,64,96]
{
    __shared__ float sK[NTOK * KV_STRIDE];          //  9216 B
    __shared__ float sV[NTOK * KV_STRIDE];          //  9216 B
    __shared__ float sP[4 * 16 * P_STRIDE];         // 17408 B

    const int bid    = blockIdx.x;                  // 0..49151
    const int h      = bid % HEADS;
    const int rem    = bid / HEADS;
    const int win    = rem % NWIN;
    const int branch = rem / NWIN;

    const float* __restrict__ src = branch ? qkv1 : qkv0;
    float* __restrict__ outp = out + (size_t)branch * ((size_t)NWIN * NTOK * DIMC);

    const int tid  = threadIdx.x;
    const int lane = tid & 31;
    const int w    = tid >> 5;        // wave 0..3
    const int l16  = lane & 15;
    const int hlf  = lane >> 4;       // 0 or 1

    const float* __restrict__ base = src + (size_t)win * (NTOK * QKV_ROW);

    // ---- cooperative load of K,V head-slices into LDS (128B-aligned chunks)
    {
        const float4* g4 = (const float4*)base;     // row = 72 float4s
        for (int i = tid; i < 512; i += 128) {
            int r = i >> 3, c4 = i & 7;
            float4 kv = g4[r * 72 + 24 + h * 8 + c4];   // k: col 96 + h*32
            float4 vv = g4[r * 72 + 48 + h * 8 + c4];   // v: col 192 + h*32
            *(float4*)(&sK[r * KV_STRIDE + c4 * 4]) = kv;
            *(float4*)(&sV[r * KV_STRIDE + c4 * 4]) = vv;
        }
    }
    __syncthreads();

    const int m0   = w * 16;
    const int mrow = m0 + l16;

    // ---- A fragments: scaled q row (global, exactly-once traffic)
    v2f afrag[8];
    {
        const float* qrow = base + mrow * QKV_ROW + h * HEAD_DIM + 2 * hlf;
        #pragma unroll
        for (int kk = 0; kk < 8; ++kk) {
            v2f t = *(const v2f*)(qrow + 4 * kk);
            afrag[kk].x = t.x * ATT_SCALE;
            afrag[kk].y = t.y * ATT_SCALE;
        }
    }

    // ---- S = (q*scale) @ k^T : four 16x16 column tiles, K=32 in 8 x4-steps
    v8f acc[4];
    #pragma unroll
    for (int nt = 0; nt < 4; ++nt) {
        v8f c = {0.f,0.f,0.f,0.f,0.f,0.f,0.f,0.f};
        const float* kb = &sK[(nt * 16 + l16) * KV_STRIDE + 2 * hlf];
        #pragma unroll
        for (int kk = 0; kk < 8; ++kk) {
            v2f b = *(const v2f*)(kb + 4 * kk);
            c = __builtin_amdgcn_wmma_f32_16x16x4_f32(
                    false, afrag[kk], false, b, (short)0, c, false, false);
        }
        acc[nt] = c;
    }

    // ---- + bias, then row softmax (normalization deferred to output)
    const float* bh = bias + h * (NTOK * NTOK);
    float rsum[8];
    #pragma unroll
    for (int j = 0; j < 8; ++j) {
        const int row = m0 + j + 8 * hlf;
        #pragma unroll
        for (int t = 0; t < 4; ++t)
            acc[t][j] += bh[row * 64 + t * 16 + l16];
        float mx = fmaxf(fmaxf(acc[0][j], acc[1][j]), fmaxf(acc[2][j], acc[3][j]));
        #pragma unroll
        for (int off = 1; off < 16; off <<= 1)
            mx = fmaxf(mx, __shfl_xor(mx, off, 32));   // stays inside 16-lane half
        float s = 0.f;
        #pragma unroll
        for (int t = 0; t < 4; ++t) {
            float e = __expf(acc[t][j] - mx);
            acc[t][j] = e;
            s += e;
        }
        #pragma unroll
        for (int off = 1; off < 16; off <<= 1)
            s += __shfl_xor(s, off, 32);
        rsum[j] = s;
    }

    // ---- stage P (C-layout -> memory) so it can be reloaded in A-layout
    float* pb = &sP[w * 16 * P_STRIDE];
    #pragma unroll
    for (int j = 0; j < 8; ++j) {
        const int lr = j + 8 * hlf;
        #pragma unroll
        for (int t = 0; t < 4; ++t)
            pb[lr * P_STRIDE + t * 16 + l16] = acc[t][j];
    }
    __syncthreads();   // uniform; also orders cross-lane LDS write->read

    // ---- A fragments of P (16 x4-chunks over K=64)
    v2f pfrag[16];
    {
        const float* pr = pb + l16 * P_STRIDE + 2 * hlf;
        #pragma unroll
        for (int kk = 0; kk < 16; ++kk)
            pfrag[kk] = *(const v2f*)(pr + 4 * kk);
    }

    // ---- O = P @ V : two 16-wide head-dim tiles
    v8f o[2];
    #pragma unroll
    for (int dt = 0; dt < 2; ++dt) {
        v8f c = {0.f,0.f,0.f,0.f,0.f,0.f,0.f,0.f};
        const float* vb = &sV[(2 * hlf) * KV_STRIDE + dt * 16 + l16];
        #pragma unroll
        for (int kk = 0; kk < 16; ++kk) {
            v2f b;
            b.x = vb[(4 * kk) * KV_STRIDE];
            b.y = vb[(4 * kk) * KV_STRIDE + KV_STRIDE];
            c = __builtin_amdgcn_wmma_f32_16x16x4_f32(
                    false, pfrag[kk], false, b, (short)0, c, false, false);
        }
        o[dt] = c;
    }

    // ---- normalize rows and store (coalesced 64B per half-wave per row)
    #pragma unroll
    for (int j = 0; j < 8; ++j) {
        const float inv = 1.0f / rsum[j];
        const int row = m0 + j + 8 * hlf;
        float* orow = outp + ((size_t)win * NTOK + row) * DIMC + h * HEAD_DIM;
        orow[l16]      = o[0][j] * inv;
        orow[16 + l16] = o[1][j] * inv;
    }
}

// ---------------------------------------------------------------------------
extern "C" void kernel_launch(void* const* d_in, const int* in_sizes, int n_in,
                              void* d_out, int out_size, void* d_ws, size_t ws_size,
                              hipStream_t stream) {
    (void)in_sizes; (void)n_in; (void)out_size; (void)ws_size;
    const float* qkv     = (const float*)d_in[0];
    const float* ass_qkv = (const float*)d_in[1];
    const float* w1      = (const float*)d_in[2];
    const float* b1      = (const float*)d_in[3];
    const float* w2      = (const float*)d_in[4];
    const float* b2      = (const float*)d_in[5];
    const float* rel_pos = (const float*)d_in[6];
    float* bias = (float*)d_ws;                 // 3*64*64 floats = 48 KB

    winattn_bias_kernel<<<16, 256, 0, stream>>>(rel_pos, w1, b1, w2, b2, bias);

    const int nblocks = 2 * NWIN * HEADS;       // 49152
    winattn_attn_kernel<<<nblocks, 128, 0, stream>>>(qkv, ass_qkv, bias, (float*)d_out);
}